// PRButterfly_22978075034044
// MI455X (gfx1250) — compile-verified
//
#include <hip/hip_runtime.h>
#include <math.h>

typedef __attribute__((ext_vector_type(2))) float v2f;
typedef __attribute__((ext_vector_type(8))) float v8f;

#define DIM 256
#define PER_STAGE 128
#define CAPV 0.99f
#define BATCH 32
#define HW 4096          // 64*64
#define TILE_P 32
#define PITCH 36         // divisible by 4 -> 16B-aligned LDS b128; 2-way conflict on column reads (cheap)

// ---------------------------------------------------------------------------
// Kernel 1: D = tanh(mod @ W.T) epilogues, via V_WMMA_F32_16X16X4_F32 (exact fp32).
// Tiles 0..63  -> w_angle (N=1024): store cos/sin of (angles[n] + tanh(acc))
// Tiles 64..79 -> w_diag  (N=256):  store clip(clip(tanh(diag[n])) + tanh(acc))
// One wave per block; EXEC all ones as WMMA requires.
// A (16x4 f32): VGPR0 = K0 (lanes 0-15) / K2 (lanes 16-31); VGPR1 = K1/K3.
// B (4x16 f32): VGPR0 = rows K0/K2 striped across lanes; VGPR1 = K1/K3.
// C/D (16x16 f32): VGPR i = row M=i (lanes 0-15), M=i+8 (lanes 16-31).
// ---------------------------------------------------------------------------
__global__ __launch_bounds__(32) void prep_kernel(
    const float* __restrict__ modulation,  // 32 x 256
    const float* __restrict__ angles,      // 1024
    const float* __restrict__ diagonal,    // 256
    const float* __restrict__ w_angle,     // 1024 x 256
    const float* __restrict__ w_diag,      // 256 x 256
    float* __restrict__ ws_cos,            // 32 x 1024
    float* __restrict__ ws_sin,            // 32 x 1024
    float* __restrict__ ws_diag)           // 32 x 256
{
    const int  tile    = blockIdx.x;
    const bool is_diag = (tile >= 64);
    const int  n0      = is_diag ? (tile - 64) * 16 : tile * 16;
    const float* W     = is_diag ? w_diag : w_angle;

    const int lane = threadIdx.x;
    const int l16  = lane & 15;
    const int hi   = lane >> 4;           // 0: K pair {0,1}, 1: K pair {2,3}

    const float* arow0 = modulation + l16 * 256;          // M tile 0: batches 0-15
    const float* arow1 = modulation + (l16 + 16) * 256;   // M tile 1: batches 16-31
    const float* wrow  = W + (n0 + l16) * 256;            // B[k][n] = W[n][k]

    v8f acc0 = {};
    v8f acc1 = {};
    for (int k0 = 0; k0 < 256; k0 += 4) {
        const int k = k0 + 2 * hi;
        v2f a0, a1, bv;
        a0.x = arow0[k];  a0.y = arow0[k + 1];
        a1.x = arow1[k];  a1.y = arow1[k + 1];
        bv.x = wrow[k];   bv.y = wrow[k + 1];
        acc0 = __builtin_amdgcn_wmma_f32_16x16x4_f32(false, a0, false, bv,
                                                     (short)0, acc0, false, false);
        acc1 = __builtin_amdgcn_wmma_f32_16x16x4_f32(false, a1, false, bv,
                                                     (short)0, acc1, false, false);
    }

    const int n = n0 + l16;
    if (!is_diag) {
        const float base = angles[n];
#pragma unroll
        for (int i = 0; i < 8; ++i) {
            const int m0 = i + 8 * hi;        // acc0 -> batches 0-15
            const int m1 = m0 + 16;           // acc1 -> batches 16-31
            float a0 = base + tanhf(acc0[i]);
            float a1 = base + tanhf(acc1[i]);
            ws_cos[m0 * 1024 + n] = cosf(a0);
            ws_sin[m0 * 1024 + n] = sinf(a0);
            ws_cos[m1 * 1024 + n] = cosf(a1);
            ws_sin[m1 * 1024 + n] = sinf(a1);
        }
    } else {
        float base = tanhf(diagonal[n]);
        base = fminf(fmaxf(base, -CAPV), CAPV);
#pragma unroll
        for (int i = 0; i < 8; ++i) {
            const int m0 = i + 8 * hi;
            const int m1 = m0 + 16;
            float d0 = base + tanhf(acc0[i]);
            float d1 = base + tanhf(acc1[i]);
            ws_diag[m0 * 256 + n] = fminf(fmaxf(d0, -CAPV), CAPV);
            ws_diag[m1 * 256 + n] = fminf(fmaxf(d1, -CAPV), CAPV);
        }
    }
}

// ---------------------------------------------------------------------------
// Kernel 2: streaming butterfly. One block = one batch x 32 spatial positions.
// b128 coalesced load (512B/wave request) -> padded LDS tile -> per-wave
// register butterfly (stages 0-4 via wave32 shfl_xor, stages 5-7 in-register)
// -> diag -> b128 coalesced store.
// ---------------------------------------------------------------------------
__global__ __launch_bounds__(256) void butterfly_kernel(
    const float* __restrict__ x,        // 32 x 256 x 4096
    const float* __restrict__ ws_cos,   // 32 x 1024
    const float* __restrict__ ws_sin,   // 32 x 1024
    const float* __restrict__ ws_diag,  // 32 x 256
    float* __restrict__ out)            // 32 x 256 x 4096
{
    __shared__ float tile[DIM * PITCH];   // 36 KB
    __shared__ float lcos[1024];
    __shared__ float lsin[1024];
    __shared__ float ldiag[DIM];

    const int b   = blockIdx.x >> 7;          // 128 tiles per batch
    const int p0  = (blockIdx.x & 127) * TILE_P;
    const int tid = threadIdx.x;
    const int wv  = tid >> 5;
    const int t   = tid & 31;

    // per-batch tables -> LDS (vectorized, coalesced)
    if (tid < 256) {
        ((float4*)lcos)[tid] = ((const float4*)(ws_cos + b * 1024))[tid];
        ((float4*)lsin)[tid] = ((const float4*)(ws_sin + b * 1024))[tid];
        ldiag[tid] = ws_diag[b * 256 + tid];
    }

    // load tile: each wave request = 4 channel rows x 32 positions (512B)
    const int cq = t >> 3;            // channel sub-row within request
    const int pq = (t & 7) * 4;       // 16B chunk within the 128B row
    const float* xb = x + ((size_t)b * DIM) * HW + p0;
#pragma unroll
    for (int cc = 0; cc < 8; ++cc) {
        const int c = wv * 32 + cc * 4 + cq;
        const float4 val = *(const float4*)(xb + (size_t)c * HW + pq);
        *(float4*)&tile[c * PITCH + pq] = val;
    }
    __syncthreads();

    // each wave owns 4 disjoint positions; lane t holds channels c = r*32 + t
    for (int j = 0; j < 4; ++j) {
        const int p = wv * 4 + j;
        float v[8];
#pragma unroll
        for (int r = 0; r < 8; ++r)
            v[r] = tile[(r * 32 + t) * PITCH + p];

        // stages 0..4: partner lane = t ^ half  (half = 1,2,4,8,16)
#pragma unroll
        for (int s = 0; s < 5; ++s) {
            const int half = 1 << s;
#pragma unroll
            for (int r = 0; r < 8; ++r) {
                const int c   = r * 32 + t;
                const int idx = s * PER_STAGE + ((c >> (s + 1)) << s) + (c & (half - 1));
                const float cv = lcos[idx];
                const float sv = lsin[idx];
                const float other = __shfl_xor(v[r], half, 32);
                const bool  right = (c >> s) & 1;
                v[r] = right ? fmaf(cv, v[r], -sv * other)
                             : fmaf(cv, v[r],  sv * other);
            }
        }
        // stages 5..7: partner register = r ^ (half/32)  (half = 32,64,128)
#pragma unroll
        for (int s = 5; s < 8; ++s) {
            const int half = 1 << s;
            const int hr   = 1 << (s - 5);
#pragma unroll
            for (int r = 0; r < 8; ++r) {
                if (r & hr) continue;
                const int rp  = r + hr;
                const int c   = r * 32 + t;
                const int idx = s * PER_STAGE + ((c >> (s + 1)) << s) + (c & (half - 1));
                const float cv = lcos[idx];
                const float sv = lsin[idx];
                const float L = v[r];
                const float R = v[rp];
                v[r]  = fmaf(cv, L,  sv * R);
                v[rp] = fmaf(cv, R, -sv * L);
            }
        }
#pragma unroll
        for (int r = 0; r < 8; ++r) {
            const int c = r * 32 + t;
            tile[c * PITCH + p] = v[r] * ldiag[c];
        }
    }
    __syncthreads();

    // coalesced b128 store
    float* ob = out + ((size_t)b * DIM) * HW + p0;
#pragma unroll
    for (int cc = 0; cc < 8; ++cc) {
        const int c = wv * 32 + cc * 4 + cq;
        *(float4*)(ob + (size_t)c * HW + pq) = *(const float4*)&tile[c * PITCH + pq];
    }
}

extern "C" void kernel_launch(void* const* d_in, const int* in_sizes, int n_in,
                              void* d_out, int out_size, void* d_ws, size_t ws_size,
                              hipStream_t stream) {
    const float* x          = (const float*)d_in[0];
    const float* modulation = (const float*)d_in[1];
    const float* angles     = (const float*)d_in[2];
    const float* diagonal   = (const float*)d_in[3];
    const float* w_angle    = (const float*)d_in[4];
    const float* w_diag     = (const float*)d_in[5];
    float* out = (float*)d_out;

    float* ws      = (float*)d_ws;
    float* ws_cos  = ws;                    // 32*1024 floats
    float* ws_sin  = ws + 32 * 1024;        // 32*1024 floats
    float* ws_diag = ws + 64 * 1024;        // 32*256  floats  (total ~288 KB)

    prep_kernel<<<80, 32, 0, stream>>>(modulation, angles, diagonal,
                                       w_angle, w_diag, ws_cos, ws_sin, ws_diag);
    butterfly_kernel<<<BATCH * (HW / TILE_P), 256, 0, stream>>>(
        x, ws_cos, ws_sin, ws_diag, out);
}